// QueryAndGroupFeature_87101936763059
// MI455X (gfx1250) — compile-verified
//
#include <hip/hip_runtime.h>

typedef __attribute__((ext_vector_type(2))) float v2f;
typedef __attribute__((ext_vector_type(4))) float v4f;
typedef __attribute__((ext_vector_type(8))) float v8f;

#define N_PTS   4096
#define NSAMPLE 32
#define C_IN    64
#define NWAVE   8                     // waves per block (split-K over refs)
#define NTILES  (N_PTS / 16)          // 256 ref tiles
#define ITERS   (NTILES / NWAVE)      // 32 ref tiles per wave
#define TPAD    20                    // tile row stride: 16B aligned, bank-safe

// ---------------------------------------------------------------------------
// Kernel 1: KNN. 8 waves per 16-query tile; each wave WMMAs (16x16x4 f32)
// distance tiles for 1/8 of the refs into a private sorted top-32, then 16
// threads do an 8-way merge per query.
//   d2 = |q|^2 - 2 q.r + |r|^2 via A=[-2qx,-2qy,-2qz,|q|^2], B=[rx;ry;rz;1],
//   C = |r|^2 broadcast  ->  one V_WMMA_F32_16X16X4_F32 per 16x16 tile.
// ---------------------------------------------------------------------------
__global__ __launch_bounds__(NWAVE * 32) void knn_wmma_kernel(const float* __restrict__ xyz,
                                                              int* __restrict__ knn_idx) {
    __shared__ __align__(16) float s_tile[NWAVE][16 * TPAD]; // padded d2 tiles
    __shared__ float s_dist[NWAVE][16][NSAMPLE + 1];         // partial sorted lists
    __shared__ int   s_idx [NWAVE][16][NSAMPLE + 1];

    const int tid  = threadIdx.x;
    const int w    = tid >> 5;            // wave 0..7
    const int lane = tid & 31;
    const int qt   = blockIdx.x;          // query tile 0..255
    const int b    = blockIdx.y;          // batch

    const float* xb = xyz + (size_t)b * N_PTS * 3;

    // init partial lists (33 entries incl. sentinel slot for merge overrun)
    for (int i = tid; i < NWAVE * 16 * (NSAMPLE + 1); i += NWAVE * 32) {
        const int ww = i / (16 * (NSAMPLE + 1));
        const int r  = i % (16 * (NSAMPLE + 1));
        s_dist[ww][r / (NSAMPLE + 1)][r % (NSAMPLE + 1)] = 3.4e38f;
        s_idx [ww][r / (NSAMPLE + 1)][r % (NSAMPLE + 1)] = 0x7fffffff;
    }
    __syncthreads();

    // A operand (16x4 f32): lanes 0-15 hold K=0,1 of row m; lanes 16-31 K=2,3
    const int half = lane >> 4;
    const int m    = lane & 15;
    const int qg   = qt * 16 + m;
    const float qx = xb[qg * 3 + 0];
    const float qy = xb[qg * 3 + 1];
    const float qz = xb[qg * 3 + 2];
    const float sq_q = qx * qx + qy * qy + qz * qz;
    v2f a;
    a.x = half ? (-2.0f * qz) : (-2.0f * qx);
    a.y = half ? sq_q         : (-2.0f * qy);

    // software-pipelined ref loads: this wave handles jt = it*8 + w
    int jt = w;
    float rx = xb[(jt * 16 + m) * 3 + 0];
    float ry = xb[(jt * 16 + m) * 3 + 1];
    float rz = xb[(jt * 16 + m) * 3 + 2];

    for (int it = 0; it < ITERS; ++it) {
        const float sq_r = rx * rx + ry * ry + rz * rz;
        v2f bb;
        bb.x = half ? rz   : rx;
        bb.y = half ? 1.0f : ry;
        v8f c;
        #pragma unroll
        for (int i = 0; i < 8; ++i) c[i] = sq_r;

        v8f d = __builtin_amdgcn_wmma_f32_16x16x4_f32(
            false, a, false, bb, (short)0, c, false, false);

        #pragma unroll
        for (int i = 0; i < 8; ++i) {
            const int row = half ? (i + 8) : i;
            s_tile[w][row * TPAD + m] = d[i];
        }

        // prefetch next ref tile before the barrier (overlaps selection)
        const int jn = jt + NWAVE;
        float nrx = 0.f, nry = 0.f, nrz = 0.f;
        if (it + 1 < ITERS) {
            nrx = xb[(jn * 16 + m) * 3 + 0];
            nry = xb[(jn * 16 + m) * 3 + 1];
            nrz = xb[(jn * 16 + m) * 3 + 2];
        }
        __syncthreads();

        // lanes 0-15 of each wave: pull the row into registers (4x b128),
        // then insert this wave's 16 candidates
        if (lane < 16) {
            const int q = lane;
            const v4f* rowp = (const v4f*)&s_tile[w][q * TPAD];
            const v4f r0 = rowp[0], r1 = rowp[1], r2 = rowp[2], r3 = rowp[3];
            float cand[16];
            #pragma unroll
            for (int i = 0; i < 4; ++i) {
                cand[i]      = r0[i];
                cand[4 + i]  = r1[i];
                cand[8 + i]  = r2[i];
                cand[12 + i] = r3[i];
            }
            float worst = s_dist[w][q][NSAMPLE - 1];
            #pragma unroll
            for (int t = 0; t < 16; ++t) {
                const float dd = cand[t];
                if (dd < worst) {
                    int p = NSAMPLE - 1;
                    #pragma clang loop unroll(disable)
                    while (p > 0 && s_dist[w][q][p - 1] > dd) {
                        s_dist[w][q][p] = s_dist[w][q][p - 1];
                        s_idx [w][q][p] = s_idx [w][q][p - 1];
                        --p;
                    }
                    s_dist[w][q][p] = dd;
                    s_idx [w][q][p] = jt * 16 + t;
                    worst = s_dist[w][q][NSAMPLE - 1];
                }
            }
        }
        __syncthreads();

        jt = jn; rx = nrx; ry = nry; rz = nrz;
    }

    // 8-way merge of partial lists; ties -> lower index (matches lax.top_k)
    if (tid < 16) {
        const int q = tid;
        int head[NWAVE];
        #pragma unroll
        for (int ww = 0; ww < NWAVE; ++ww) head[ww] = 0;
        int* orow = knn_idx + ((size_t)b * N_PTS + (size_t)(qt * 16 + q)) * NSAMPLE;
        for (int k = 0; k < NSAMPLE; ++k) {
            float bestd = 3.5e38f;
            int   besti = 0x7fffffff;
            int   bestw = 0;
            #pragma unroll
            for (int ww = 0; ww < NWAVE; ++ww) {
                const float dv = s_dist[ww][q][head[ww]];
                const int   iv = s_idx [ww][q][head[ww]];
                const bool better = (dv < bestd) || (dv == bestd && iv < besti);
                if (better) { bestd = dv; besti = iv; bestw = ww; }
            }
            orow[k] = besti;
            #pragma unroll
            for (int ww = 0; ww < NWAVE; ++ww) head[ww] += (ww == bestw) ? 1 : 0;
        }
    }
}

// ---------------------------------------------------------------------------
// Kernel 2: gather + relative-diff + concat. Output-stream bound (256 MB, NT
// stores since out > L2); features (1 MB/batch) stay L2-resident for gathers.
// ---------------------------------------------------------------------------
__global__ __launch_bounds__(256) void group_feature_kernel(const float* __restrict__ features,
                                                            const int* __restrict__ knn_idx,
                                                            float* __restrict__ out) {
    const int m  = blockIdx.x;
    const int b  = blockIdx.y;
    const int k  = threadIdx.x & 31;     // neighbor slot (coalesced 128B stores)
    const int cr = threadIdx.x >> 5;     // channel lane 0..7

    const int id = knn_idx[((size_t)b * N_PTS + m) * NSAMPLE + k];

    const float* fb = features + (size_t)b * C_IN * N_PTS;
    float* ob = out + (size_t)b * (2 * C_IN) * (size_t)N_PTS * NSAMPLE;

    #pragma unroll
    for (int cc = 0; cc < C_IN / 8; ++cc) {
        const int c = cc * 8 + cr;
        const float center = fb[(size_t)c * N_PTS + m];
        const float g      = fb[(size_t)c * N_PTS + id];
        const size_t o1 = ((size_t)c * N_PTS + m) * NSAMPLE + k;
        const size_t o2 = ((size_t)(c + C_IN) * N_PTS + m) * NSAMPLE + k;
        __builtin_nontemporal_store(g - center, &ob[o1]);
        __builtin_nontemporal_store(g,          &ob[o2]);
    }
}

// ---------------------------------------------------------------------------
extern "C" void kernel_launch(void* const* d_in, const int* in_sizes, int n_in,
                              void* d_out, int out_size, void* d_ws, size_t ws_size,
                              hipStream_t stream) {
    const float* xyz      = (const float*)d_in[0];   // (b, 4096, 3)
    const float* features = (const float*)d_in[1];   // (b, 64, 4096)
    float* out            = (float*)d_out;           // (b, 128, 4096, 32)
    int* knn_idx          = (int*)d_ws;              // (b, 4096, 32) = 2 MB scratch

    const int B = in_sizes[0] / (N_PTS * 3);

    knn_wmma_kernel<<<dim3(NTILES, B), NWAVE * 32, 0, stream>>>(xyz, knn_idx);
    group_feature_kernel<<<dim3(N_PTS, B), 256, 0, stream>>>(features, knn_idx, out);
}